// GSpadeNodeEmbedding_21277267984961
// MI455X (gfx1250) — compile-verified
//
#include <hip/hip_runtime.h>
#include <math.h>

// Problem constants (from reference)
#define T_LEN  32768
#define N_SEG  2048
#define DSEQ   128
#define DL     512      // D_LSTM
#define DX     512
#define DPROJ  1024
#define DHID   2048

typedef float v2f __attribute__((ext_vector_type(2)));
typedef float v8f __attribute__((ext_vector_type(8)));

// D(16x16,f32) += A(16x4,f32) x B(4x16,f32)  -- full fp32 precision path
__device__ __forceinline__ v8f wmma4(v2f a, v2f b, v8f c) {
  return __builtin_amdgcn_wmma_f32_16x16x4_f32(
      /*neg_a=*/false, a, /*neg_b=*/false, b,
      /*c_mod=*/(short)0, c, /*reuse_a=*/false, /*reuse_b=*/false);
}

// ---------------------------------------------------------------------------
// Kernel 1: segment bounds from sorted masks.  Every value 0..N-1 appears.
// ---------------------------------------------------------------------------
__global__ void seg_bounds_kernel(const int* __restrict__ masks,
                                  int* __restrict__ seg_start,
                                  int* __restrict__ seg_end) {
  int t = blockIdx.x * blockDim.x + threadIdx.x;
  if (t >= T_LEN) return;
  int m = masks[t];
  if (t == 0 || masks[t - 1] != m) seg_start[m] = t;
  if (t == T_LEN - 1 || masks[t + 1] != m) seg_end[m] = t + 1;
}

// ---------------------------------------------------------------------------
// Kernel 2: repack a weight matrix W[nout][K] (row-major, used as B^T) into
// WMMA-B fragment order:  dst[((nt*(K/4)+k0q)*32 + lane)*2 + sub]
// where lane = 16*(k%4 >> 1) + (n%16), sub = k&1.  After repacking, one
// wave's B fragment per WMMA is a single contiguous 256B burst.
// ---------------------------------------------------------------------------
__global__ void repack_b_kernel(const float* __restrict__ src,
                                float* __restrict__ dst, int nout, int K) {
  int idx = blockIdx.x * blockDim.x + threadIdx.x;
  if (idx >= nout * K) return;
  int n = idx / K, k = idx % K;
  int nt = n >> 4, frm = n & 15;
  int k0q = k >> 2, rem = k & 3;
  int lane = ((rem >> 1) << 4) | frm;
  int sub = rem & 1;
  dst[((((size_t)nt * (K >> 2) + k0q) << 5) + lane) * 2 + sub] = src[idx];
}

// ---------------------------------------------------------------------------
// Kernel 3: out[:, 0:1024] = x @ Wx^T + bx  (2048x512x1024 WMMA GEMM)
// grid (128 row-bands, 8 col-groups); block stages 16 x-rows to LDS once,
// 8 waves each own one 16-col tile; B from fragment-repacked Wx.
// ---------------------------------------------------------------------------
__global__ void __launch_bounds__(256, 1)
proj_gemm_kernel(const float* __restrict__ x, const float* __restrict__ Wx_p,
                 const float* __restrict__ bx, float* __restrict__ out) {
  const int tid  = threadIdx.x;
  const int lane = tid & 31;
  const int wv   = tid >> 5;
  const int row0 = blockIdx.x * 16;
  const int nt   = blockIdx.y * 8 + wv;            // 64 n-tiles total

  __shared__ float Xs[16][516];                    // pitch 516: conflict-free

  {  // coalesced stage: each thread copies 32 contiguous floats of one row
    int m = tid >> 4, part = tid & 15;
    const float4* src = (const float4*)(x + (size_t)(row0 + m) * DX + part * 32);
    float4* dstp = (float4*)&Xs[m][part * 32];
#pragma unroll
    for (int q = 0; q < 8; ++q) dstp[q] = src[q];
  }
  __syncthreads();

  const int frm = lane & 15;
  const int fkb = (lane >> 4) << 1;
  const int hi8 = (lane >> 4) << 3;

  const float* ap = &Xs[frm][0];
  const float* bp = Wx_p + ((size_t)nt * (DX >> 2) << 5) * 2 + lane * 2;
  v8f acc = {};
  for (int k0 = 0; k0 < DX; k0 += 4) {
    v2f a = *(const v2f*)(ap + k0 + fkb);
    v2f b = *(const v2f*)(bp + (k0 >> 2) * 64);    // contiguous per wave
    acc = wmma4(a, b, acc);
  }
  const float bv = bx[nt * 16 + frm];
#pragma unroll
  for (int r = 0; r < 8; ++r)
    out[(size_t)(row0 + r + hi8) * DHID + nt * 16 + frm] = acc[r] + bv;
}

// ---------------------------------------------------------------------------
// Kernel 4: fused segment RNN scan + mean pool.
// grid = (128 groups of 16 segments, 2 directions), 256 threads = 8 waves.
// Per step k: H_new = tanh(S_k @ Wih^T + b + [k>0] * H @ Whh^T)  (16x512)
// 32 column tiles, 4 per wave; A from LDS, B from L2-resident repacked W.
// Pooled sums accumulate in registers (fixed lane<->element mapping).
// ---------------------------------------------------------------------------
__global__ void __launch_bounds__(256, 1)
scan_pool_kernel(const float* __restrict__ seqs,
                 const float* __restrict__ WihF_p, const float* __restrict__ WhhF_p,
                 const float* __restrict__ bias_f,
                 const float* __restrict__ WihB_p, const float* __restrict__ WhhB_p,
                 const float* __restrict__ bias_b,
                 const int* __restrict__ seg_start, const int* __restrict__ seg_end,
                 float* __restrict__ out) {
  const int dir = blockIdx.y;
  const int s0  = blockIdx.x * 16;
  const float* Wih  = dir ? WihB_p : WihF_p;
  const float* Whh  = dir ? WhhB_p : WhhF_p;
  const float* bias = dir ? bias_b : bias_f;

  __shared__ float S[16][132];     // staged seq rows
  __shared__ float H[16][516];     // recurrent state (conflict-free pitch)
  __shared__ int   sstart[16], slen[16];
  __shared__ int   skmax;

  const int tid  = threadIdx.x;
  const int lane = tid & 31;
  const int wv   = tid >> 5;
  const int frm  = lane & 15;
  const int fkb  = (lane >> 4) << 1;
  const int hi8  = (lane >> 4) << 3;

  if (tid < 16) {
    int st = seg_start[s0 + tid];
    sstart[tid] = st;
    slen[tid]   = seg_end[s0 + tid] - st;
  }
  __syncthreads();
  if (tid == 0) {
    int km = 0;
    for (int i = 0; i < 16; ++i) km = max(km, slen[i]);
    skmax = km;
  }
  __syncthreads();
  const int kmax = skmax;

  int lenreg[8];
#pragma unroll
  for (int r = 0; r < 8; ++r) lenreg[r] = slen[r + hi8];

  float biasreg[4];
#pragma unroll
  for (int i = 0; i < 4; ++i) biasreg[i] = bias[(wv * 4 + i) * 16 + frm];

  // fragment-order B base pointers for this wave's 4 column tiles
  const float* bih[4];
  const float* bhh[4];
#pragma unroll
  for (int i = 0; i < 4; ++i) {
    int tj = wv * 4 + i;
    bih[i] = Wih + ((size_t)tj * (DSEQ >> 2) << 5) * 2 + lane * 2;
    bhh[i] = Whh + ((size_t)tj * (DL   >> 2) << 5) * 2 + lane * 2;
  }

  float pacc[4][8];
#pragma unroll
  for (int i = 0; i < 4; ++i)
#pragma unroll
    for (int r = 0; r < 8; ++r) pacc[i][r] = 0.f;
  float hreg[4][8];

  for (int k = 0; k < kmax; ++k) {
    // ---- stage the 16 gathered seq rows for this step into LDS ----
    {
      int m = tid >> 4, part = tid & 15;
      int len = slen[m], st = sstart[m];
      int t = (k < len) ? (dir ? (st + len - 1 - k) : (st + k)) : st;
      const float4* src = (const float4*)(seqs + (size_t)t * DSEQ + part * 8);
      float4 a0 = src[0], a1 = src[1];
      float* drow = &S[m][part * 8];
      ((float4*)drow)[0] = a0;
      ((float4*)drow)[1] = a1;
    }
    __syncthreads();

    // ---- WMMA compute: 4 column tiles per wave ----
    for (int i = 0; i < 4; ++i) {
      v8f acc = {};
      const float* ap = &S[frm][0];
      const float* bp = bih[i];
      __builtin_prefetch(bhh[i], 0, 0);
      for (int k0 = 0; k0 < DSEQ; k0 += 4) {
        v2f a = *(const v2f*)(ap + k0 + fkb);
        v2f b = *(const v2f*)(bp + (k0 >> 2) * 64);
        acc = wmma4(a, b, acc);
      }
      // recurrent term (start gate is uniform per step: off at k==0)
      if (k > 0) {
        const float* ah = &H[frm][0];
        const float* bh = bhh[i];
        for (int k0 = 0; k0 < DL; k0 += 4) {
          v2f a = *(const v2f*)(ah + k0 + fkb);
          v2f b = *(const v2f*)(bh + (k0 >> 2) * 64);
          acc = wmma4(a, b, acc);
        }
      }
#pragma unroll
      for (int r = 0; r < 8; ++r) hreg[i][r] = tanhf(acc[r] + biasreg[i]);
    }
    __syncthreads();   // all reads of H complete before overwriting

    // ---- write H back + masked pooled accumulation ----
#pragma unroll
    for (int i = 0; i < 4; ++i) {
      const int col0 = (wv * 4 + i) * 16;
#pragma unroll
      for (int r = 0; r < 8; ++r) {
        int m = r + hi8;
        float h = hreg[i][r];
        H[m][col0 + frm] = h;
        if (k < lenreg[r]) pacc[i][r] += h;
      }
    }
    __syncthreads();   // H visible before next step's reads
  }

  // ---- pooled mean -> out[:, 1024 + dir*512 + col] ----
#pragma unroll
  for (int i = 0; i < 4; ++i) {
    const int col0 = (wv * 4 + i) * 16;
#pragma unroll
    for (int r = 0; r < 8; ++r) {
      int m = r + hi8;
      float inv = 1.0f / (float)lenreg[r];
      out[(size_t)(s0 + m) * DHID + DPROJ + dir * DL + col0 + frm] =
          pacc[i][r] * inv;
    }
  }
}

// ---------------------------------------------------------------------------
extern "C" void kernel_launch(void* const* d_in, const int* in_sizes, int n_in,
                              void* d_out, int out_size, void* d_ws, size_t ws_size,
                              hipStream_t stream) {
  const float* x      = (const float*)d_in[0];
  const float* seqs   = (const float*)d_in[1];
  const int*   masks  = (const int*)  d_in[2];
  const float* Wih_f  = (const float*)d_in[3];
  const float* Whh_f  = (const float*)d_in[4];
  const float* b_f    = (const float*)d_in[5];
  const float* Wih_b  = (const float*)d_in[6];
  const float* Whh_b  = (const float*)d_in[7];
  const float* b_b    = (const float*)d_in[8];
  const float* Wx     = (const float*)d_in[9];
  const float* bx     = (const float*)d_in[10];
  float* out = (float*)d_out;

  // workspace layout
  int*   seg_start = (int*)d_ws;                       // N_SEG ints
  int*   seg_end   = seg_start + N_SEG;                // N_SEG ints
  float* wsf       = (float*)(seg_end + N_SEG);
  float* Wx_p   = wsf;                                 // 1024*512
  float* WihF_p = Wx_p   + (size_t)DPROJ * DX;         // 512*128
  float* WhhF_p = WihF_p + (size_t)DL * DSEQ;          // 512*512
  float* WihB_p = WhhF_p + (size_t)DL * DL;            // 512*128
  float* WhhB_p = WihB_p + (size_t)DL * DSEQ;          // 512*512

  seg_bounds_kernel<<<T_LEN / 256, 256, 0, stream>>>(masks, seg_start, seg_end);

  // one-time fragment-order weight repacks (all stay L2-resident: ~4.5 MB)
  repack_b_kernel<<<(DPROJ * DX) / 256, 256, 0, stream>>>(Wx,    Wx_p,   DPROJ, DX);
  repack_b_kernel<<<(DL * DSEQ) / 256, 256, 0, stream>>>(Wih_f, WihF_p, DL, DSEQ);
  repack_b_kernel<<<(DL * DL)   / 256, 256, 0, stream>>>(Whh_f, WhhF_p, DL, DL);
  repack_b_kernel<<<(DL * DSEQ) / 256, 256, 0, stream>>>(Wih_b, WihB_p, DL, DSEQ);
  repack_b_kernel<<<(DL * DL)   / 256, 256, 0, stream>>>(Whh_b, WhhB_p, DL, DL);

  // 128 row-bands x 8 col-groups; A staged in LDS, shared by 8 waves
  proj_gemm_kernel<<<dim3(128, 8), 256, 0, stream>>>(x, Wx_p, bx, out);

  scan_pool_kernel<<<dim3(N_SEG / 16, 2), 256, 0, stream>>>(
      seqs, WihF_p, WhhF_p, b_f, WihB_p, WhhB_p, b_b, seg_start, seg_end, out);
}